// ChebConv_86517821211314
// MI455X (gfx1250) — compile-verified
//
#include <hip/hip_runtime.h>
#include <hip/hip_bf16.h>

typedef float v2f __attribute__((ext_vector_type(2)));
typedef float v8f __attribute__((ext_vector_type(8)));

#define VV    50000      // vertices
#define BB    4          // batch
#define FIN   128
#define FOUT  128
#define KCH   5          // Chebyshev order
#define CCOL  (FIN*BB)   // 512 columns of each x_k buffer, col = f*4 + b

// ---------------------------------------------------------------------------
// x0[v*512 + f*4 + b] = inputs[b, v, f].  Thread t = v*128 + f: reads four
// coalesced batch streams, writes one coalesced float4.
// ---------------------------------------------------------------------------
__global__ __launch_bounds__(256) void build_x0_kernel(
    const float* __restrict__ in, float* __restrict__ x0) {
  int t = blockIdx.x * 256 + threadIdx.x;       // 0 .. V*128-1
  int v = t >> 7;
  int f = t & 127;
  float4 o;
  o.x = in[((size_t)0 * VV + v) * FIN + f];
  o.y = in[((size_t)1 * VV + v) * FIN + f];
  o.z = in[((size_t)2 * VV + v) * FIN + f];
  o.w = in[((size_t)3 * VV + v) * FIN + f];
  ((float4*)x0)[t] = o;                          // x0[v*512 + f*4 + 0..3]
}

// ============================ CSR construction =============================
__global__ __launch_bounds__(256) void zero_counts_kernel(int* __restrict__ p) {
  int i = blockIdx.x * 256 + threadIdx.x;
  if (i < VV) p[i] = 0;
}

__global__ __launch_bounds__(256) void count_rows_kernel(
    const int* __restrict__ rows, int* __restrict__ counts, int nnz) {
  int i = blockIdx.x * 256 + threadIdx.x;
  if (i < nnz) atomicAdd(&counts[rows[i]], 1);
}

// single-workgroup exclusive scan over counts[V] -> row_start[V+1], cursor[V]
__global__ __launch_bounds__(1024) void scan_kernel(
    const int* __restrict__ counts, int* __restrict__ row_start,
    int* __restrict__ cursor) {
  __shared__ int sdata[1024];
  __shared__ int soffset;
  int tid = threadIdx.x;
  if (tid == 0) soffset = 0;
  __syncthreads();
  for (int base = 0; base < VV; base += 1024) {
    int idx = base + tid;
    int v = (idx < VV) ? counts[idx] : 0;
    sdata[tid] = v;
    __syncthreads();
    for (int off = 1; off < 1024; off <<= 1) {   // Hillis-Steele inclusive scan
      int t = (tid >= off) ? sdata[tid - off] : 0;
      __syncthreads();
      sdata[tid] += t;
      __syncthreads();
    }
    int incl = sdata[tid];
    int off0 = soffset;                          // safe: last write was pre-barrier
    if (idx < VV) {
      int excl = off0 + incl - v;
      row_start[idx] = excl;
      cursor[idx] = excl;
    }
    __syncthreads();
    if (tid == 1023) soffset = off0 + incl;
    __syncthreads();
  }
  if (tid == 0) row_start[VV] = soffset;
}

__global__ __launch_bounds__(256) void scatter_edges_kernel(
    const int* __restrict__ rows, const int* __restrict__ cols,
    const float* __restrict__ vals, int* __restrict__ cursor,
    int* __restrict__ ccols, float* __restrict__ cvals, int nnz) {
  int i = blockIdx.x * 256 + threadIdx.x;
  if (i >= nnz) return;
  int p = atomicAdd(&cursor[rows[i]], 1);
  ccols[p] = cols[i];
  cvals[p] = vals[i];
}

// ---------------------------------------------------------------------------
// Atomic-free CSR SpMM fused with the Chebyshev combine:
//   y[r,:] = alpha * y[r,:] + scale * sum_e vals[e] * x[cols[e],:]
// One wave per row (8 rows / 256-thread block); 16 fp32 accumulators per lane
// live in VGPRs. Edge (col,val) batches are loaded coalesced and broadcast
// with __shfl (ds_bpermute on wave32). In-place over y is safe: only row r of
// y is read (for alpha != 0) before row r is written; gathers read only x.
// ---------------------------------------------------------------------------
__global__ __launch_bounds__(256) void spmm_csr_kernel(
    float* __restrict__ y, const float* __restrict__ x,
    const int* __restrict__ row_start, const int* __restrict__ ccols,
    const float* __restrict__ cvals, float alpha, float scale) {
  int wave = threadIdx.x >> 5;
  int lane = threadIdx.x & 31;
  int r = blockIdx.x * 8 + wave;
  if (r >= VV) return;

  float4* __restrict__ yr = (float4*)(y + (size_t)r * CCOL);
  float4 acc[4];
  if (alpha == 0.0f) {
#pragma unroll
    for (int j = 0; j < 4; ++j) acc[j] = make_float4(0.f, 0.f, 0.f, 0.f);
  } else {
#pragma unroll
    for (int j = 0; j < 4; ++j) {
      float4 t = yr[lane + 32 * j];
      acc[j] = make_float4(alpha * t.x, alpha * t.y, alpha * t.z, alpha * t.w);
    }
  }

  int e0 = row_start[r];
  int e1 = row_start[r + 1];
  for (int base = e0; base < e1; base += 32) {
    int n = e1 - base;
    if (n > 32) n = 32;
    int cv = 0;
    float wv = 0.0f;
    if (lane < n) {
      cv = ccols[base + lane];
      wv = cvals[base + lane] * scale;
    }
    for (int i = 0; i < n; ++i) {
      int c = __shfl(cv, i, 32);
      float w = __shfl(wv, i, 32);
      const float4* __restrict__ xs = (const float4*)(x + (size_t)c * CCOL);
#pragma unroll
      for (int j = 0; j < 4; ++j) {
        float4 t = xs[lane + 32 * j];
        acc[j].x += w * t.x;
        acc[j].y += w * t.y;
        acc[j].z += w * t.z;
        acc[j].w += w * t.w;
      }
    }
  }

#pragma unroll
  for (int j = 0; j < 4; ++j) yr[lane + 32 * j] = acc[j];
}

// ---------------------------------------------------------------------------
// Fused per-k GEMM accumulation with V_WMMA_F32_16X16X4_F32 (fp32-exact):
//   out[b*V+v, n] (+)= sum_f x[v*512 + f*4 + b] * W[(f*5+k)*128 + n]
// One block = 16-v stripe x 4 batches x 128 cols. 8 waves: wave -> (batch,
// N-half); each wave runs 4 independent 16x16 accumulator chains, 128 WMMAs.
// ---------------------------------------------------------------------------
__global__ __launch_bounds__(256) void gemm_pass_kernel(
    float* __restrict__ out,        // [B*V, 128]
    const float* __restrict__ x,    // [V, 512] current x_k
    const float* __restrict__ w,    // flat [640*128]
    const float* __restrict__ bias, // [128]
    int k, int first) {
  __shared__ float lA[BB][16][132];

  int tid = threadIdx.x;
  int v0 = blockIdx.x * 16;

  for (int i = tid; i < 16 * 128; i += 256) {
    int m = i >> 7;
    int f = i & 127;
    float4 t4 = *(const float4*)(x + (size_t)(v0 + m) * CCOL + f * 4);
    lA[0][m][f] = t4.x;
    lA[1][m][f] = t4.y;
    lA[2][m][f] = t4.z;
    lA[3][m][f] = t4.w;
  }
  __syncthreads();

  int wave = tid >> 5;
  int lane = tid & 31;
  int bsel = wave & 3;
  int nh   = wave >> 2;
  int lm   = lane & 15;
  int hi   = lane >> 4;
  int ncol = nh * 64 + lm;

  v8f c[4];
  if (first) {
#pragma unroll
    for (int t = 0; t < 4; ++t) {
      float bv = bias[ncol + t * 16];
#pragma unroll
      for (int j = 0; j < 8; ++j) c[t][j] = bv;
    }
  } else {
#pragma unroll
    for (int t = 0; t < 4; ++t)
#pragma unroll
      for (int j = 0; j < 8; ++j)
        c[t][j] = out[(size_t)(bsel * VV + v0 + j + hi * 8) * FOUT + ncol + t * 16];
  }

  const float* wp = w + (size_t)(2 * hi * KCH + k) * FOUT + ncol;
#pragma unroll 8
  for (int f0 = 0; f0 < 128; f0 += 4) {
    int fa = f0 + 2 * hi;
    v2f a;
    a[0] = lA[bsel][lm][fa];
    a[1] = lA[bsel][lm][fa + 1];
#pragma unroll
    for (int t = 0; t < 4; ++t) {
      v2f bf;
      bf[0] = wp[t * 16];
      bf[1] = wp[t * 16 + KCH * FOUT];
      c[t] = __builtin_amdgcn_wmma_f32_16x16x4_f32(
          false, a, false, bf, (short)0, c[t], false, false);
    }
    wp += 4 * KCH * FOUT;
  }

#pragma unroll
  for (int t = 0; t < 4; ++t)
#pragma unroll
    for (int j = 0; j < 8; ++j)
      out[(size_t)(bsel * VV + v0 + j + hi * 8) * FOUT + ncol + t * 16] = c[t][j];
}

// ---------------------------------------------------------------------------
extern "C" void kernel_launch(void* const* d_in, const int* in_sizes, int n_in,
                              void* d_out, int out_size, void* d_ws, size_t ws_size,
                              hipStream_t stream) {
  const int*   rows   = (const int*)d_in[0];
  const int*   cols   = (const int*)d_in[1];
  const float* vals   = (const float*)d_in[2];
  const float* inputs = (const float*)d_in[3];
  const float* weight = (const float*)d_in[4];
  const float* bias   = (const float*)d_in[5];
  float* out = (float*)d_out;
  int nnz = in_sizes[0];

  // workspace layout:
  //   bufA, bufB : 2 x V*512 floats (ping-pong x_k)          204.8 MB
  //   counts, cursor, row_start : V, V, V+1 ints              ~0.6 MB
  //   ccols (nnz ints), cvals (nnz floats)                    ~12.8 MB
  float* bufA = (float*)d_ws;
  float* bufB = bufA + (size_t)VV * CCOL;
  int*   counts    = (int*)(bufB + (size_t)VV * CCOL);
  int*   cursor    = counts + VV;
  int*   row_start = cursor + VV;
  int*   ccols     = row_start + (VV + 1) + 1;   // +1 keeps 8B alignment slack
  float* cvals     = (float*)(ccols + nnz);

  const int bx0Grid  = (VV * FIN) / 256;        // 25000
  const int edgeGrid = (nnz + 255) / 256;
  const int vtxGrid  = (VV + 255) / 256;
  const int spmmGrid = (VV + 7) / 8;            // 6250
  const int gemmGrid = VV / 16;                 // 3125

  // ---- build CSR (once per call; only cheap int atomics) ----
  zero_counts_kernel<<<vtxGrid, 256, 0, stream>>>(counts);
  count_rows_kernel<<<edgeGrid, 256, 0, stream>>>(rows, counts, nnz);
  scan_kernel<<<1, 1024, 0, stream>>>(counts, row_start, cursor);
  scatter_edges_kernel<<<edgeGrid, 256, 0, stream>>>(rows, cols, vals, cursor,
                                                     ccols, cvals, nnz);

  // ---- x0 ; out = bias + A0 @ W0 ----
  build_x0_kernel<<<bx0Grid, 256, 0, stream>>>(inputs, bufA);
  gemm_pass_kernel<<<gemmGrid, 256, 0, stream>>>(out, bufA, weight, bias, 0, 1);

  // ---- x1 = L x0 ; out += A1 @ W1 ----
  spmm_csr_kernel<<<spmmGrid, 256, 0, stream>>>(bufB, bufA, row_start, ccols,
                                                cvals, 0.0f, 1.0f);
  gemm_pass_kernel<<<gemmGrid, 256, 0, stream>>>(out, bufB, weight, bias, 1, 0);

  // ---- x2 = 2 L x1 - x0 (in place over x0) ; out += A2 @ W2 ----
  spmm_csr_kernel<<<spmmGrid, 256, 0, stream>>>(bufA, bufB, row_start, ccols,
                                                cvals, -1.0f, 2.0f);
  gemm_pass_kernel<<<gemmGrid, 256, 0, stream>>>(out, bufA, weight, bias, 2, 0);

  // ---- x3 = 2 L x2 - x1 ; out += A3 @ W3 ----
  spmm_csr_kernel<<<spmmGrid, 256, 0, stream>>>(bufB, bufA, row_start, ccols,
                                                cvals, -1.0f, 2.0f);
  gemm_pass_kernel<<<gemmGrid, 256, 0, stream>>>(out, bufB, weight, bias, 3, 0);

  // ---- x4 = 2 L x3 - x2 ; out += A4 @ W4 ----
  spmm_csr_kernel<<<spmmGrid, 256, 0, stream>>>(bufA, bufB, row_start, ccols,
                                                cvals, -1.0f, 2.0f);
  gemm_pass_kernel<<<gemmGrid, 256, 0, stream>>>(out, bufA, weight, bias, 4, 0);
}